// HOR_18897856102484
// MI455X (gfx1250) — compile-verified
//
#include <hip/hip_runtime.h>
#include <hip/hip_bf16.h>

// Problem constants (from reference setup_inputs)
#define BB 8
#define CH 160
#define CL 80
#define NN 4096   // H*W = 64*64

typedef __attribute__((ext_vector_type(2))) float v2f;
typedef __attribute__((ext_vector_type(8))) float v8f;

// ---------------------------------------------------------------------------
// Kernel 1: fold the linear chain  Wc = Wh @ W1,  bc = Wh @ b1 + bh
// ---------------------------------------------------------------------------
__global__ void k_combine(const float* __restrict__ W1, const float* __restrict__ b1,
                          const float* __restrict__ Wh, const float* __restrict__ bh,
                          float* __restrict__ Wc, float* __restrict__ bc) {
    int idx = blockIdx.x * blockDim.x + threadIdx.x;
    if (idx < CL * CH) {
        int o = idx / CH, i = idx - o * CH;
        float s = 0.f;
        for (int j = 0; j < CH; ++j) s += Wh[o * CH + j] * W1[j * CH + i];
        Wc[idx] = s;
    } else if (idx < CL * CH + CL) {
        int o = idx - CL * CH;
        float s = bh[o];
        for (int j = 0; j < CH; ++j) s += Wh[o * CH + j] * b1[j];
        bc[o] = s;
    }
}

// ---------------------------------------------------------------------------
// Kernel 2: projections.  blockIdx.y selects matrix:
//   0: XL = Wc @ x_latter + bc   (written straight into d_out second half)
//   1: XQ = Wl @ x + bl
//   2: V  = Wv @ x + bv
// ---------------------------------------------------------------------------
__global__ void __launch_bounds__(256)
k_proj(const float* __restrict__ xlat, const float* __restrict__ x,
       const float* __restrict__ Wc, const float* __restrict__ bc,
       const float* __restrict__ Wl, const float* __restrict__ bl,
       const float* __restrict__ Wv, const float* __restrict__ bv,
       float* __restrict__ XL, float* __restrict__ XQ, float* __restrict__ V) {
    int idx = blockIdx.x * blockDim.x + threadIdx.x;   // over BB*CL*NN
    int n = idx & (NN - 1);
    int c = (idx >> 12) % CL;
    int b = idx / (CL * NN);
    const float *W, *bias, *src;
    float* dst;
    int Cin;
    if (blockIdx.y == 0)      { W = Wc; bias = bc; src = xlat + (size_t)b * CH * NN; Cin = CH; dst = XL; }
    else if (blockIdx.y == 1) { W = Wl; bias = bl; src = x    + (size_t)b * CL * NN; Cin = CL; dst = XQ; }
    else                      { W = Wv; bias = bv; src = x    + (size_t)b * CL * NN; Cin = CL; dst = V;  }
    float s = bias[c];
    const float* wr = W + c * Cin;
    for (int i = 0; i < Cin; ++i) s += wr[i] * src[(size_t)i * NN + n];
    dst[idx] = s;
}

// ---------------------------------------------------------------------------
// Kernel 3: per-column softmax stats (softmax over axis=1 == over n, per m).
// Block = 8 waves * 16 columns = 128 columns of m.  The XQ n-tile (16x80) is
// shared by all 8 waves: double-buffered in LDS, filled with the CDNA5
// async-copy path (global_load_async_to_lds_b32, ASYNCcnt) overlapped with
// the WMMA compute of the current tile.  A-fragments then come from LDS as
// aligned ds_load_b64 ([n][stride 82] layout: (c,c+1) contiguous, 8B aligned,
// stride mod 64 = 18 -> conflict-free across 16 lanes).
// ---------------------------------------------------------------------------
#define AST 82   // LDS row stride (floats) for staged A tile
__global__ void __launch_bounds__(256)
k_stats(const float* __restrict__ XQ, const float* __restrict__ XL,
        float* __restrict__ Ms, float* __restrict__ Ss) {
    __shared__ float a_lds[2][16 * AST];

    const int tid  = threadIdx.x;
    const int lane = tid & 31;
    const int wave = tid >> 5;
    const int b    = blockIdx.y;
    const int m0   = blockIdx.x * 128 + wave * 16;
    const int ln   = lane & 15;
    const int hi2  = (lane >> 4) << 1;

    const float* q  = XQ + (size_t)b * CL * NN;
    const float* km = XL + (size_t)b * CL * NN;

    // this thread's share of the cooperative tile copy: 5 dwords
    const int cc[5] = { (tid) >> 4, (tid + 256) >> 4, (tid + 512) >> 4,
                        (tid + 768) >> 4, (tid + 1024) >> 4 };
    const int nl = tid & 15;

    // per-wave B fragments (XL columns m0..m0+15), resident for whole kernel
    v2f bf[20];
#pragma unroll
    for (int k = 0; k < 20; ++k) {
        int c = 4 * k + hi2;
        bf[k].x = km[(size_t)c * NN + m0 + ln];
        bf[k].y = km[(size_t)(c + 1) * NN + m0 + ln];
    }

    // prologue: async-fill buffer 0 with n-tile 0
#pragma unroll
    for (int j = 0; j < 5; ++j) {
        unsigned loff = (unsigned)(uintptr_t)&a_lds[0][nl * AST + cc[j]];
        const float* g = q + (size_t)cc[j] * NN + nl;
        asm volatile("global_load_async_to_lds_b32 %0, %1, off"
                     :: "v"(loff), "v"(g) : "memory");
    }
    asm volatile("s_wait_asynccnt 0x0" ::: "memory");
    __syncthreads();

    float Mrun = -3.0e38f, Srun = 0.f;
    for (int it = 0; it < NN / 16; ++it) {
        const int cur = it & 1;
        // prefetch next tile into the other buffer (overlaps with compute)
        if (it + 1 < NN / 16) {
            const int n1 = (it + 1) * 16;
#pragma unroll
            for (int j = 0; j < 5; ++j) {
                unsigned loff = (unsigned)(uintptr_t)&a_lds[cur ^ 1][nl * AST + cc[j]];
                const float* g = q + (size_t)cc[j] * NN + n1 + nl;
                asm volatile("global_load_async_to_lds_b32 %0, %1, off"
                             :: "v"(loff), "v"(g) : "memory");
            }
        }
        // ---- logits tile from staged A + resident B ----
        v8f acc = {};
#pragma unroll
        for (int k = 0; k < 20; ++k) {
            v2f a = *(const v2f*)&a_lds[cur][ln * AST + 4 * k + hi2];  // ds_load_b64
            acc = __builtin_amdgcn_wmma_f32_16x16x4_f32(false, a, false, bf[k],
                                                        (short)0, acc, false, false);
        }
        // ---- online column max/sum; lane pair (l, l+16) holds one column ----
        float tmax = acc[0];
#pragma unroll
        for (int r = 1; r < 8; ++r) tmax = fmaxf(tmax, acc[r]);
        tmax = fmaxf(tmax, __shfl_xor(tmax, 16, 32));
        float Mnew = fmaxf(Mrun, tmax);
        float tsum = 0.f;
#pragma unroll
        for (int r = 0; r < 8; ++r) tsum += __expf(acc[r] - Mnew);
        tsum += __shfl_xor(tsum, 16, 32);
        Srun = Srun * __expf(Mrun - Mnew) + tsum;
        Mrun = Mnew;

        asm volatile("s_wait_asynccnt 0x0" ::: "memory");
        __syncthreads();
    }
    if (lane < 16) {
        Ms[(size_t)b * NN + m0 + lane] = Mrun;
        Ss[(size_t)b * NN + m0 + lane] = Srun;
    }
}

// ---------------------------------------------------------------------------
// Kernel 4: e_out[b][c][n] = sum_m P[n,m] * V[c,m] + XQ[c,n]
// Block = 4 waves, a 32-row n band (two 16-row tiles per wave); wave w covers
// m in [w*1024,(w+1)*1024).  Every XL / V B-fragment load is shared by both
// n-tiles (halves B traffic per WMMA).  P is restaged through per-wave LDS
// (stride 18 -> aligned ds_load_b64 A-fragments).  Cross-wave reduce via
// ds_add_f32 LDS atomics, then add XQ residual and store [B,C,N].
// ---------------------------------------------------------------------------
#define PST 18   // LDS row stride (floats) for staged P tiles
__global__ void __launch_bounds__(128)
k_out(const float* __restrict__ XQ, const float* __restrict__ XL,
      const float* __restrict__ V, const float* __restrict__ Ms,
      const float* __restrict__ Ss, float* __restrict__ Eout) {
    __shared__ float E_lds[32 * CL];
    __shared__ float P_lds[4][2][16 * PST];

    const int tid  = threadIdx.x;
    const int lane = tid & 31;
    const int wave = tid >> 5;
    const int b    = blockIdx.y;
    const int n0   = blockIdx.x * 32;
    const int ln   = lane & 15;
    const int hi2  = (lane >> 4) << 1;
    const int rhi  = (lane >> 4) << 3;

    for (int i = tid; i < 32 * CL; i += 128) E_lds[i] = 0.f;
    __syncthreads();

    const float* q  = XQ + (size_t)b * CL * NN;
    const float* km = XL + (size_t)b * CL * NN;
    const float* vm = V  + (size_t)b * CL * NN;
    const float* Mp = Ms + (size_t)b * NN;
    const float* Sp = Ss + (size_t)b * NN;

    // A-fragments of XQ for both n-tiles (reused for all m-tiles)
    v2f aq0[20], aq1[20];
#pragma unroll
    for (int k = 0; k < 20; ++k) {
        int c = 4 * k + hi2;
        aq0[k].x = q[(size_t)c * NN + n0 + ln];
        aq0[k].y = q[(size_t)(c + 1) * NN + n0 + ln];
        aq1[k].x = q[(size_t)c * NN + n0 + 16 + ln];
        aq1[k].y = q[(size_t)(c + 1) * NN + n0 + 16 + ln];
    }

    v8f E0[5] = {}, E1[5] = {};
    float* pl0 = &P_lds[wave][0][0];
    float* pl1 = &P_lds[wave][1][0];

    for (int mt = 0; mt < 64; ++mt) {
        const int m0 = wave * 1024 + mt * 16;
        // ---- two logits tiles, B fragments loaded once ----
        v8f acc0 = {}, acc1 = {};
#pragma unroll
        for (int k = 0; k < 20; ++k) {
            int c = 4 * k + hi2;
            v2f bfr;
            bfr.x = km[(size_t)c * NN + m0 + ln];
            bfr.y = km[(size_t)(c + 1) * NN + m0 + ln];
            acc0 = __builtin_amdgcn_wmma_f32_16x16x4_f32(false, aq0[k], false, bfr,
                                                         (short)0, acc0, false, false);
            acc1 = __builtin_amdgcn_wmma_f32_16x16x4_f32(false, aq1[k], false, bfr,
                                                         (short)0, acc1, false, false);
        }
        // ---- P = exp(L - M[m]) / S[m]; stage to LDS (D-layout -> row-major) ----
        float Mc = Mp[m0 + ln];
        float rS = __builtin_amdgcn_rcpf(Sp[m0 + ln]);
#pragma unroll
        for (int r = 0; r < 8; ++r) {
            pl0[(r + rhi) * PST + ln] = __expf(acc0[r] - Mc) * rS;
            pl1[(r + rhi) * PST + ln] = __expf(acc1[r] - Mc) * rS;
        }
        // ---- E += P @ V^T, V fragments shared by both n-tiles ----
#pragma unroll
        for (int kk = 0; kk < 4; ++kk) {
            const int mk = 4 * kk + hi2;
            v2f ap0 = *(const v2f*)&pl0[ln * PST + mk];   // ds_load_b64
            v2f ap1 = *(const v2f*)&pl1[ln * PST + mk];
#pragma unroll
            for (int ct = 0; ct < 5; ++ct) {
                v2f bv;
                size_t base = (size_t)(ct * 16 + ln) * NN + m0 + mk;
                bv.x = vm[base];
                bv.y = vm[base + 1];
                E0[ct] = __builtin_amdgcn_wmma_f32_16x16x4_f32(false, ap0, false, bv,
                                                               (short)0, E0[ct], false, false);
                E1[ct] = __builtin_amdgcn_wmma_f32_16x16x4_f32(false, ap1, false, bv,
                                                               (short)0, E1[ct], false, false);
            }
        }
    }

    // ---- cross-wave reduce into E_lds[n][c] via ds_add_f32 ----
#pragma unroll
    for (int ct = 0; ct < 5; ++ct)
#pragma unroll
        for (int r = 0; r < 8; ++r) {
            int n = r + rhi;
            int c = ct * 16 + ln;
            atomicAdd(&E_lds[n * CL + c], E0[ct][r]);
            atomicAdd(&E_lds[(n + 16) * CL + c], E1[ct][r]);
        }
    __syncthreads();

    // ---- add XQ residual (e += xq), store transposed layout [B][C][N] ----
    for (int i = tid; i < 32 * CL; i += 128) {
        int n = i & 31;
        int c = i >> 5;
        float val = E_lds[n * CL + c] + q[(size_t)c * NN + n0 + n];
        Eout[(size_t)(b * CL + c) * NN + n0 + n] = val;
    }
}

// ---------------------------------------------------------------------------
extern "C" void kernel_launch(void* const* d_in, const int* in_sizes, int n_in,
                              void* d_out, int out_size, void* d_ws, size_t ws_size,
                              hipStream_t stream) {
    const float* x_latter = (const float*)d_in[0];
    const float* x        = (const float*)d_in[1];
    const float* W1 = (const float*)d_in[2];
    const float* b1 = (const float*)d_in[3];
    const float* Wh = (const float*)d_in[4];
    const float* bh = (const float*)d_in[5];
    const float* Wl = (const float*)d_in[6];
    const float* bl = (const float*)d_in[7];
    const float* Wv = (const float*)d_in[8];
    const float* bv = (const float*)d_in[9];

    float* out    = (float*)d_out;
    float* e_out  = out;                              // output 0: e  [B,C,N]
    float* xl_out = out + (size_t)BB * CL * NN;       // output 1: xl [B,C,N]

    // workspace layout (floats): Wc | bc | XQ | V | Ms | Ss  (~21.3 MB)
    float* ws = (float*)d_ws;
    float* Wc = ws;
    float* bc = Wc + CL * CH;
    float* XQ = bc + CL;
    float* V  = XQ + (size_t)BB * CL * NN;
    float* Ms = V  + (size_t)BB * CL * NN;
    float* Ss = Ms + (size_t)BB * NN;

    k_combine<<<dim3((CL * CH + CL + 255) / 256), dim3(256), 0, stream>>>(W1, b1, Wh, bh, Wc, bc);
    k_proj<<<dim3((BB * CL * NN) / 256, 3), dim3(256), 0, stream>>>(
        x_latter, x, Wc, bc, Wl, bl, Wv, bv, xl_out, XQ, V);
    k_stats<<<dim3(NN / 128, BB), dim3(256), 0, stream>>>(XQ, xl_out, Ms, Ss);
    k_out<<<dim3(NN / 32, BB), dim3(128), 0, stream>>>(XQ, xl_out, V, Ms, Ss, e_out);
}